// CommunityTrustGNN_80023830659561
// MI455X (gfx1250) — compile-verified
//
#include <hip/hip_runtime.h>

#define D1 64   // input / hidden feature width (layer-1 K, layer-1 N, layer-2 K)
#define D2 64
#define D3 32   // layer-2 output width

typedef __attribute__((ext_vector_type(2))) float v2f;
typedef __attribute__((ext_vector_type(8))) float v8f;

// ---------------------------------------------------------------- zero fill
__global__ void zero_f32_kernel(float* __restrict__ p, long n) {
    long i = (long)blockIdx.x * blockDim.x + threadIdx.x;
    long stride = (long)gridDim.x * blockDim.x;
    for (; i < n; i += stride) p[i] = 0.0f;
}

// ------------------------------------------------- edge-parallel scatter-add
// thread = edge * 16 + chunk; each thread moves one float4 of the 64-wide row.
// agg[dst] += feat[src]; optionally deg[dst] += 1.
__global__ void scatter_accum_kernel(const float* __restrict__ feat,
                                     const long long* __restrict__ src,
                                     const long long* __restrict__ dst,
                                     float* __restrict__ agg,
                                     float* __restrict__ deg,
                                     int nEdges, int addDeg) {
    const int chunks = D1 / 4;  // 16
    long tid = (long)blockIdx.x * blockDim.x + threadIdx.x;
    long total = (long)nEdges * chunks;
    if (tid >= total) return;
    int e = (int)(tid / chunks);
    int c = (int)(tid % chunks);
    int s = (int)src[e];
    int d = (int)dst[e];
    const float4 v = *(const float4*)(feat + (long)s * D1 + c * 4);
    float* p = agg + (long)d * D1 + c * 4;
    atomicAdd(p + 0, v.x);
    atomicAdd(p + 1, v.y);
    atomicAdd(p + 2, v.z);
    atomicAdd(p + 3, v.w);
    if (addDeg && c == 0) atomicAdd(deg + d, 1.0f);
}

// ------------------------------------------------------- deg -> 1/max(deg,1)
__global__ void invdeg_kernel(float* __restrict__ deg, int n) {
    int i = blockIdx.x * blockDim.x + threadIdx.x;
    if (i < n) deg[i] = 1.0f / fmaxf(deg[i], 1.0f);
}

// ---------------------------------------------------------------- WMMA GEMM
// out[M x N] = act( (aggsum .* invdeg_row) @ Wl + xin @ Wr + bias )
// One 16x16 output tile per wave32, fp32 WMMA (V_WMMA_F32_16X16X4_F32).
// Fragment layouts (ISA 7.12.2):
//   A 16x4 f32 : lane&15 = M row, K = k + vgpr + 2*(lane>=16)  (even float2)
//   B 4x16 f32 : lane&15 = N col, K = k + vgpr + 2*(lane>=16)
//   C/D 16x16  : lane&15 = N col, M = vgpr + 8*(lane>=16)
template <int K, int N, bool RELU>
__global__ void sage_gemm_wmma_kernel(const float* __restrict__ aggsum,
                                      const float* __restrict__ invdeg,
                                      const float* __restrict__ xin,
                                      const float* __restrict__ Wl,
                                      const float* __restrict__ Wr,
                                      const float* __restrict__ bias,
                                      float* __restrict__ out, int M) {
    const int NT = N >> 4;
    int wave = blockIdx.x * (blockDim.x >> 5) + (threadIdx.x >> 5);
    int lane = threadIdx.x & 31;
    int totalTiles = (M >> 4) * NT;
    if (wave >= totalTiles) return;  // wave-uniform: EXEC stays all-ones

    int mt = wave / NT, nt = wave % NT;
    int h = lane >> 4, l = lane & 15;
    int m = (mt << 4) + l;     // A row this lane feeds
    int n = (nt << 4) + l;     // B/D column this lane holds
    float scale = invdeg[m];

    const float* arow = aggsum + (long)m * K;
    const float* xrow = xin + (long)m * K;

    v8f c = {0.f, 0.f, 0.f, 0.f, 0.f, 0.f, 0.f, 0.f};

    // chain 1: mean-aggregated neighbors @ Wl
#pragma unroll
    for (int k = 0; k < K; k += 4) {
        int kk = k + (h << 1);
        float2 av = *(const float2*)(arow + kk);
        v2f a; a.x = av.x * scale; a.y = av.y * scale;
        v2f b; b.x = Wl[kk * N + n]; b.y = Wl[(kk + 1) * N + n];
        c = __builtin_amdgcn_wmma_f32_16x16x4_f32(false, a, false, b,
                                                  (short)0, c, false, false);
    }
    // chain 2: root features @ Wr (accumulate into same C)
#pragma unroll
    for (int k = 0; k < K; k += 4) {
        int kk = k + (h << 1);
        float2 xv = *(const float2*)(xrow + kk);
        v2f a; a.x = xv.x; a.y = xv.y;
        v2f b; b.x = Wr[kk * N + n]; b.y = Wr[(kk + 1) * N + n];
        c = __builtin_amdgcn_wmma_f32_16x16x4_f32(false, a, false, b,
                                                  (short)0, c, false, false);
    }

    float bn = bias[n];
#pragma unroll
    for (int r = 0; r < 8; r++) {
        float v = c[r] + bn;
        if (RELU) v = fmaxf(v, 0.0f);
        out[(long)((mt << 4) + r + (h << 3)) * N + n] = v;
    }
}

// ------------------------------------------------------------- trust head
// t = relu(h2 @ Wt1 + bt1); trust = sigmoid(t @ Wt2 + bt2). One node/thread.
__global__ void trust_head_kernel(const float* __restrict__ h2,
                                  const float* __restrict__ Wt1,
                                  const float* __restrict__ bt1,
                                  const float* __restrict__ Wt2,
                                  const float* __restrict__ bt2,
                                  float* __restrict__ out, int M) {
    int nd = blockIdx.x * blockDim.x + threadIdx.x;
    if (nd >= M) return;
    const float* row = h2 + (long)nd * D3;
    float hbuf[D3];
#pragma unroll
    for (int i = 0; i < D3; i++) hbuf[i] = row[i];
    float z = bt2[0];
#pragma unroll
    for (int j = 0; j < 16; j++) {
        float t = bt1[j];
#pragma unroll
        for (int i = 0; i < D3; i++) t = fmaf(hbuf[i], Wt1[i * 16 + j], t);
        t = fmaxf(t, 0.0f);
        z = fmaf(t, Wt2[j], z);
    }
    out[nd] = 1.0f / (1.0f + expf(-z));
}

// ----------------------------------------------------------------- launcher
extern "C" void kernel_launch(void* const* d_in, const int* in_sizes, int n_in,
                              void* d_out, int out_size, void* d_ws, size_t ws_size,
                              hipStream_t stream) {
    const float* x       = (const float*)d_in[0];
    const long long* ei  = (const long long*)d_in[1];  // int64 [2, E]
    const float* Wl1 = (const float*)d_in[2];
    const float* Wr1 = (const float*)d_in[3];
    const float* b1  = (const float*)d_in[4];
    const float* Wl2 = (const float*)d_in[5];
    const float* Wr2 = (const float*)d_in[6];
    const float* b2  = (const float*)d_in[7];
    const float* Wt1 = (const float*)d_in[8];
    const float* bt1 = (const float*)d_in[9];
    const float* Wt2 = (const float*)d_in[10];
    const float* bt2 = (const float*)d_in[11];

    const int nNodes = in_sizes[0] / D1;   // 100000
    const int nEdges = in_sizes[1] / 2;    // 1600000
    const long long* src = ei;
    const long long* dst = ei + nEdges;

    float* ws  = (float*)d_ws;
    float* agg = ws;                              // nNodes*64 floats
    float* deg = agg + (size_t)nNodes * D1;       // nNodes (becomes inv-deg)
    float* h1  = deg + nNodes;                    // nNodes*64 floats
    float* h2    = (float*)d_out;                 // nNodes*32 (output h)
    float* trust = h2 + (size_t)nNodes * D3;      // nNodes   (output trust)

    // 1) zero agg + deg (contiguous region)
    zero_f32_kernel<<<2048, 256, 0, stream>>>(agg, (long)nNodes * (D1 + 1));

    // 2) layer-1 scatter: agg[dst] += x[src], deg[dst] += 1
    {
        long total = (long)nEdges * (D1 / 4);
        int blocks = (int)((total + 255) / 256);
        scatter_accum_kernel<<<blocks, 256, 0, stream>>>(x, src, dst, agg, deg,
                                                         nEdges, 1);
    }

    // 3) deg -> 1/max(deg,1)
    invdeg_kernel<<<(nNodes + 255) / 256, 256, 0, stream>>>(deg, nNodes);

    // 4) layer-1 GEMM (fp32 WMMA) + ReLU -> h1
    {
        int tiles = (nNodes / 16) * (D2 / 16);
        int blocks = (tiles + 7) / 8;  // 8 waves per 256-thread block
        sage_gemm_wmma_kernel<D1, D2, true>
            <<<blocks, 256, 0, stream>>>(agg, deg, x, Wl1, Wr1, b1, h1, nNodes);
    }

    // 5) re-zero agg (keep inv-deg)
    zero_f32_kernel<<<2048, 256, 0, stream>>>(agg, (long)nNodes * D1);

    // 6) layer-2 scatter: agg[dst] += h1[src]
    {
        long total = (long)nEdges * (D1 / 4);
        int blocks = (int)((total + 255) / 256);
        scatter_accum_kernel<<<blocks, 256, 0, stream>>>(h1, src, dst, agg, deg,
                                                         nEdges, 0);
    }

    // 7) layer-2 GEMM (fp32 WMMA), no activation -> h2 (d_out)
    {
        int tiles = (nNodes / 16) * (D3 / 16);
        int blocks = (tiles + 7) / 8;
        sage_gemm_wmma_kernel<D2, D3, false>
            <<<blocks, 256, 0, stream>>>(agg, deg, h1, Wl2, Wr2, b2, h2, nNodes);
    }

    // 8) trust head -> d_out tail
    trust_head_kernel<<<(nNodes + 255) / 256, 256, 0, stream>>>(
        h2, Wt1, bt1, Wt2, bt2, trust, nNodes);
}